// RegionLoss_23398981829294
// MI455X (gfx1250) — compile-verified
//
#include <hip/hip_runtime.h>
#include <cmath>

#define NB   32
#define NA   5
#define NC   80
#define NH   64
#define NW   64
#define MAXT 50

typedef float v2f __attribute__((ext_vector_type(2)));
typedef float v8f __attribute__((ext_vector_type(8)));

// ANCHORS_PX pairs / STRIDE(32): (42.3,55.4),(102.2,128.3),(161.8,259.2),(303.1,154.9),(359.6,320.2)
__device__ __constant__ float c_aw[NA] = {42.3f/32.f, 102.2f/32.f, 161.8f/32.f, 303.1f/32.f, 359.6f/32.f};
__device__ __constant__ float c_ah[NA] = {55.4f/32.f, 128.3f/32.f, 259.2f/32.f, 154.9f/32.f, 320.2f/32.f};

// sigmoid with v_rcp_f32 instead of the IEEE divide sequence (div_fmas/div_fixup)
__device__ __forceinline__ float sigmoidf_(float v) {
    return __builtin_amdgcn_rcpf(1.0f + __expf(-v));
}

__global__ void rl_zero(float* loss) {
    if (threadIdx.x == 0 && blockIdx.x == 0) loss[0] = 0.0f;
}

// Per-target preprocessing: one block per batch, thread t handles target t.
// Record layout (16 floats): [gx1, gx2, gy1, gy2, garea, key(-1 if invalid),
//                             tx, ty, tw, th, t_iou, gcls, 0,0,0,0]
__global__ void rl_prep(const float* __restrict__ out,
                        const float* __restrict__ tgt,
                        float* __restrict__ recs) {
    __shared__ float nz[MAXT];
    const int b = blockIdx.x;
    const int t = threadIdx.x;
    if (t < MAXT) nz[t] = (tgt[b*MAXT*5 + t*5 + 1] != 0.0f) ? 1.0f : 0.0f;
    __syncthreads();
    if (t >= MAXT) return;

    bool valid = true;
    for (int s = 0; s <= t; ++s) valid = valid && (nz[s] != 0.0f);   // cumprod prefix

    const float* tg = tgt + b*MAXT*5 + t*5;
    const float gcls = tg[0];
    const float gx = tg[1] * (float)NW, gy = tg[2] * (float)NH;
    const float gw = tg[3] * (float)NW, gh = tg[4] * (float)NH;

    // best anchor by wh-IoU (argmax, first max wins -> strict >)
    float best = -1.0f; int bn = 0;
    for (int n = 0; n < NA; ++n) {
        const float aw = c_aw[n], ah = c_ah[n];
        const float inter = fminf(gw, aw) * fminf(gh, ah);
        const float uni   = gw*gh + aw*ah - inter;
        const float r     = inter * __builtin_amdgcn_rcpf(uni);
        if (r > best) { best = r; bn = n; }
    }
    const int gi = min(max((int)gx, 0), NW - 1);
    const int gj = min(max((int)gy, 0), NH - 1);
    const float aw = c_aw[bn], ah = c_ah[bn];

    // gather predicted box at (b, bn, gj, gi) for t_iou
    const size_t base = ((size_t)(b*NA + bn) * (5 + NC)) * (size_t)(NH*NW) + (size_t)(gj*NW + gi);
    const float ox = out[base];
    const float oy = out[base + (size_t)(NH*NW)];
    const float ow = out[base + (size_t)(2*NH*NW)];
    const float oh = out[base + (size_t)(3*NH*NW)];
    const float pbx = sigmoidf_(ox) + (float)gi;
    const float pby = sigmoidf_(oy) + (float)gj;
    const float pbw = __expf(ow) * aw;
    const float pbh = __expf(oh) * ah;
    const float iw = fminf(pbx + 0.5f*pbw, gx + 0.5f*gw) - fmaxf(pbx - 0.5f*pbw, gx - 0.5f*gw);
    const float ih = fminf(pby + 0.5f*pbh, gy + 0.5f*gh) - fmaxf(pby - 0.5f*pbh, gy - 0.5f*gh);
    const float inter = fmaxf(iw, 0.0f) * fmaxf(ih, 0.0f);
    const float tiou  = inter * __builtin_amdgcn_rcpf(pbw*pbh + gw*gh - inter);

    float* r = recs + (size_t)(b*MAXT + t) * 16;
    r[0]  = gx - 0.5f*gw;  r[1] = gx + 0.5f*gw;
    r[2]  = gy - 0.5f*gh;  r[3] = gy + 0.5f*gh;
    r[4]  = gw * gh;
    r[5]  = valid ? (float)((bn*NH + gj)*NW + gi) : -1.0f;  // <= 20479: exact in fp32
    r[6]  = gx - (float)gi;
    r[7]  = gy - (float)gj;
    r[8]  = __logf(gw / aw);
    r[9]  = __logf(gh / ah);
    r[10] = tiou;
    r[11] = gcls;
    r[12] = 0.0f; r[13] = 0.0f; r[14] = 0.0f; r[15] = 0.0f;
}

// Main fused pass: one thread per (b,a,j,i) cell. Grid exactly tiles all
// cells (2560 blocks x 256), so EXEC is all-ones at the WMMA reduction.
__global__ void rl_main(const float* __restrict__ out,
                        const float* __restrict__ recs,
                        float* __restrict__ loss) {
    __shared__ float srec[MAXT * 16];
    const int tid  = threadIdx.x;
    const int cell = blockIdx.x * 256 + tid;
    const int b    = cell / (NA*NH*NW);            // each block stays in one batch (20480 % 256 == 0)

    for (int k = tid; k < MAXT*16; k += 256)
        srec[k] = recs[(size_t)b * (MAXT*16) + k];
    __syncthreads();

    const int rem = cell % (NA*NH*NW);
    const int a   = rem >> 12;                     // / 4096
    const int jw  = rem & 4095;
    const int i   = jw & 63;
    const int j   = jw >> 6;

    const size_t base = ((size_t)(b*NA + a) * (5 + NC)) * (size_t)(NH*NW) + (size_t)jw;
    const float ox = out[base];
    const float oy = out[base + 4096];
    const float ow = out[base + 2*4096];
    const float oh = out[base + 3*4096];
    const float oc = out[base + 4*4096];

    const float x    = sigmoidf_(ox);
    const float y    = sigmoidf_(oy);
    const float conf = sigmoidf_(oc);
    const float pw   = __expf(ow) * c_aw[a];
    const float ph   = __expf(oh) * c_ah[a];
    const float px   = x + (float)i, py = y + (float)j;
    const float px1 = px - 0.5f*pw, px2 = px + 0.5f*pw;
    const float py1 = py - 0.5f*ph, py2 = py + 0.5f*ph;
    const float parea = pw * ph;
    const float cellkey = (float)rem;              // (a*NH + j)*NW + i == rem

    int  match = -1;
    bool over  = false;
    for (int t = 0; t < MAXT; ++t) {
        const float* r = &srec[t*16];
        const float key = r[5];
        if (key < 0.0f) break;                     // validity is a prefix property
        const float iw = fminf(px2, r[1]) - fmaxf(px1, r[0]);
        const float ih = fminf(py2, r[3]) - fmaxf(py1, r[2]);
        const float inter = fmaxf(iw, 0.0f) * fmaxf(ih, 0.0f);
        // iou > 0.6  <=>  inter > 0.6*(parea + garea - inter)  (division-free)
        over = over || (inter > 0.6f * (parea + r[4] - inter));
        if (key == cellkey) match = t;             // last write wins
    }

    float txg = 0.5f, tyg = 0.5f, twg = 0.0f, thg = 0.0f;
    float cmask = over ? 0.0f : 1.0f;              // NOOBJECT_SCALE = 1
    float tconf = 0.0f, lcls = 0.0f;

    if (match >= 0) {
        const float* r = &srec[match*16];
        txg = r[6]; tyg = r[7]; twg = r[8]; thg = r[9];
        cmask = 5.0f;                              // OBJECT_SCALE
        tconf = r[10];
        const int cid = (int)r[11];
        const size_t cbase = base + (size_t)5*4096;
        float m = -3.0e38f;
        for (int c = 0; c < NC; ++c) m = fmaxf(m, out[cbase + (size_t)c*4096]);
        float s = 0.0f;
        for (int c = 0; c < NC; ++c) s += __expf(out[cbase + (size_t)c*4096] - m);
        const float logZ = m + __logf(s);
        lcls = logZ - out[cbase + (size_t)cid*4096];
    }

    const float dx = x - txg, dy = y - tyg, dw = ow - twg, dh = oh - thg, dc = conf - tconf;
    float lsum = 0.5f*(dx*dx + dy*dy + dw*dw + dh*dh + cmask*dc*dc) + lcls;

    // ---- wave32 reduction via the matrix unit ----
    // A (16x4 f32, 2 VGPRs): lane L -> A[L%16, L<16 ? 0 : 2] = lsum, K=1,3 zero.
    // B = ones(4x16)  =>  D[m,n] = lsum[m] + lsum[m+16]  (independent of n).
    v2f A;  A.x = lsum; A.y = 0.0f;
    v2f Bm; Bm.x = 1.0f; Bm.y = 1.0f;
    v8f C = {};
    v8f D = __builtin_amdgcn_wmma_f32_16x16x4_f32(
        false, A, false, Bm, (short)0, C, false, false);
    // lane n holds D[0..7 (+8 for lanes>=16), n]; sum its 8 rows, then pair lanes.
    float s = D[0] + D[1] + D[2] + D[3] + D[4] + D[5] + D[6] + D[7];
    s += __shfl_xor(s, 16);                        // now every lane has the wave total
    if ((tid & 31) == 0) atomicAdd(loss, s);
}

extern "C" void kernel_launch(void* const* d_in, const int* in_sizes, int n_in,
                              void* d_out, int out_size, void* d_ws, size_t ws_size,
                              hipStream_t stream) {
    const float* d_output = (const float*)d_in[0];   // (32, 425, 64, 64) f32
    const float* d_target = (const float*)d_in[1];   // (32, 250) f32
    float* loss = (float*)d_out;                     // scalar f32
    float* recs = (float*)d_ws;                      // NB*MAXT*16 floats = 100 KB

    rl_zero<<<1, 1, 0, stream>>>(loss);
    rl_prep<<<NB, 64, 0, stream>>>(d_output, d_target, recs);
    rl_main<<<(NB*NA*NH*NW) / 256, 256, 0, stream>>>(d_output, recs, loss);
}